// DP_Discriminator_43009802502448
// MI455X (gfx1250) — compile-verified
//
#include <hip/hip_runtime.h>
#include <cmath>

// ---------------------------------------------------------------------------
// GRU (2 layers, B=128, T=1024, H=256) + FC + sigmoid for MI455X (gfx1250).
// fp32 math via V_WMMA_F32_16X16X4_F32 (wave32); async global->LDS staging of
// the per-step gate activations in the sequential scan.
// ---------------------------------------------------------------------------

typedef float v2f __attribute__((ext_vector_type(2)));
typedef float v8f __attribute__((ext_vector_type(8)));
typedef int   v4i __attribute__((ext_vector_type(4)));

#define AS1 __attribute__((address_space(1)))
#define AS3 __attribute__((address_space(3)))

#define H_DIM   256
#define G3      768          // 3*H
#define T_LEN   1024
#define B_SZ    128
#define LDS_STR 260          // h-tile row stride: 256 + 4 pad (bank spread, f4 align)
#define XS_STR  772          // xg-tile row stride: 768 + 4 pad (768%64==0 otherwise)

// D = A(16x4) x B(4x16) + C, fp32.  Lane mapping (ISA 7.12.2, 32-bit):
//   A: m = lane&15, k = 2*(lane>>4)+{0,1};  B: n = lane&15, k = 2*(lane>>4)+{0,1}
//   C/D: n = lane&15, m = v + 8*(lane>>4) for VGPR v in 0..7
static __device__ __forceinline__ v8f wmma_f32(v2f a, v2f b, v8f c) {
    return __builtin_amdgcn_wmma_f32_16x16x4_f32(false, a, false, b, (short)0, c,
                                                 false, false);
}

// Async 16-byte global -> LDS copy (ASYNCcnt-tracked, bypasses VGPRs).
// Builtin signature (from hipcc diagnostic): (int4 AS1* src, int4 AS3* dst,
// imm offset, imm cpol).
static __device__ __forceinline__ void async_b128(const float* g, float* l) {
#if __has_builtin(__builtin_amdgcn_global_load_async_to_lds_b128)
    __builtin_amdgcn_global_load_async_to_lds_b128(
        (AS1 v4i*)g, (AS3 v4i*)l, 0, 0);
#else
    unsigned loff = (unsigned)(size_t)(AS3 v4i*)l;
    asm volatile("global_load_async_to_lds_b128 %0, %1, off"
                 :: "v"(loff), "v"(g) : "memory");
#endif
}

#if __has_builtin(__builtin_amdgcn_s_wait_asynccnt)
#define WAIT_ASYNC(n) __builtin_amdgcn_s_wait_asynccnt(n)
#else
#define WAIT_ASYNC(n) asm volatile("s_wait_asynccnt %0" :: "i"(n) : "memory")
#endif

// ---------------------------------------------------------------------------
// Phase 1: XG = seq @ W^T + bias      seq:(B*T,256) W:(768,256) -> XG:(B*T,768)
// grid = (B*T/16, 6), block = 256 (8 waves); wave w does N-tile blockIdx.y*8+w.
// ---------------------------------------------------------------------------
__global__ __launch_bounds__(256)
void xg_gemm_kernel(const float* __restrict__ seq,
                    const float* __restrict__ W,
                    const float* __restrict__ bias,
                    float* __restrict__ xg)
{
    __shared__ float Atile[16 * LDS_STR];
    const int tid = threadIdx.x;
    const int m0  = blockIdx.x * 16;

    for (int i = tid * 4; i < 16 * 256; i += 256 * 4) {
        const int r = i >> 8, c = i & 255;
        const float4 v = *(const float4*)(seq + (size_t)(m0 + r) * H_DIM + c);
        *(float4*)(&Atile[r * LDS_STR + c]) = v;
    }
    __syncthreads();

    const int wv   = tid >> 5;
    const int lane = tid & 31;
    const int lm   = lane & 15;
    const int lk2  = (lane >> 4) * 2;
    const int n0   = (blockIdx.y * 8 + wv) * 16;

    const float* arow = &Atile[lm * LDS_STR + lk2];
    const float* brow = W + (size_t)(n0 + lm) * H_DIM + lk2;   // B[k][n] = W[n][k]

    v8f acc = {};
#pragma unroll 8
    for (int k = 0; k < H_DIM; k += 4) {
        v2f a = *(const v2f*)(arow + k);
        v2f b = *(const v2f*)(brow + k);
        acc = wmma_f32(a, b, acc);
    }

    const float bn = bias[n0 + lm];
    const int   mb = (lane >> 4) * 8;
#pragma unroll
    for (int v = 0; v < 8; ++v)   // streamed once, >L2: non-temporal store
        __builtin_nontemporal_store(acc[v] + bn,
                                    &xg[(size_t)(m0 + mb + v) * G3 + n0 + lm]);
}

// ---------------------------------------------------------------------------
// Phase 2: sequential GRU scan. One WG per 16 batch rows (row-independent
// recurrence -> no cross-WG sync). 16 waves; wave w owns hidden cols
// [w*16, w*16+16) and its r/z/n gate tiles. xg slice (48KB/step) is
// double-buffered in LDS via async global->LDS DMA, overlapped with WMMA.
// ---------------------------------------------------------------------------
__global__ __launch_bounds__(512)
void gru_scan_kernel(const float* __restrict__ xg,    // (B*T, 768), incl. b_ih
                     const float* __restrict__ Whh,   // (768, 256)
                     const float* __restrict__ bhh,   // (768)
                     float* __restrict__ out_seq)     // (B, T, 256)
{
    __shared__ float Hs[16 * LDS_STR];          // 16.6 KB  h state
    __shared__ float Xs[2 * 16 * XS_STR];       // 98.8 KB  xg double buffer

    const int tid = threadIdx.x;
    const int b0  = blockIdx.x * 16;

    for (int i = tid; i < 16 * LDS_STR; i += 512) Hs[i] = 0.0f;

    const int wv   = tid >> 5;
    const int lane = tid & 31;
    const int lm   = lane & 15;
    const int lk2  = (lane >> 4) * 2;
    const int mb   = (lane >> 4) * 8;
    const int col  = wv * 16 + lm;          // hidden column this lane produces

    const float br  = bhh[col];
    const float bz  = bhh[H_DIM + col];
    const float bnn = bhh[2 * H_DIM + col];

    const float* wr = Whh + (size_t)(0 * H_DIM + col) * H_DIM + lk2;
    const float* wz = Whh + (size_t)(1 * H_DIM + col) * H_DIM + lk2;
    const float* wn = Whh + (size_t)(2 * H_DIM + col) * H_DIM + lk2;

    float hreg[8];
#pragma unroll
    for (int v = 0; v < 8; ++v) hreg[v] = 0.0f;

    // Wave w DMAs batch-row w of a timestep: 768 floats = 6 x (32 lanes x 16B).
    const int c0 = lane * 4;
    const float* grow = xg + ((size_t)(b0 + wv) * T_LEN) * G3 + c0;

    // Prefill buffer 0 with t=0.
#pragma unroll
    for (int i = 0; i < 6; ++i)
        async_b128(grow + i * 128, &Xs[wv * XS_STR + c0 + i * 128]);

    __syncthreads();   // Hs zero-init visible

    for (int t = 0; t < T_LEN; ++t) {
        // Kick off DMA for t+1 into the other buffer (overlaps the k-loop).
        const int     tn = (t + 1 < T_LEN) ? (t + 1) : (T_LEN - 1);
        const int     bn = (t + 1) & 1;
        const float*  gs = grow + (size_t)tn * G3;
        float*        ls = &Xs[(bn * 16 + wv) * XS_STR + c0];
#pragma unroll
        for (int i = 0; i < 6; ++i)
            async_b128(gs + i * 128, ls + i * 128);

        // hg = h @ Whh^T for this wave's 3 gate tiles, shared A fragment.
        v8f ar = {}, az = {}, an = {};
        const float* hrow = &Hs[lm * LDS_STR + lk2];
#pragma unroll 4
        for (int k = 0; k < H_DIM; k += 4) {
            v2f a   = *(const v2f*)(hrow + k);
            v2f b0f = *(const v2f*)(wr + k);
            v2f b1f = *(const v2f*)(wz + k);
            v2f b2f = *(const v2f*)(wn + k);
            ar = wmma_f32(a, b0f, ar);
            az = wmma_f32(a, b1f, az);
            an = wmma_f32(a, b2f, an);
        }

        // Async loads complete in order: allowing the 6 newest (t+1) to stay
        // outstanding guarantees this step's buffer is fully in LDS.
        WAIT_ASYNC(6);
        __syncthreads();   // all waves' DMA rows + Hs reads complete

        const float* xb = &Xs[((t & 1) * 16) * XS_STR];
#pragma unroll
        for (int v = 0; v < 8; ++v) {
            const int m = mb + v;
            const float xr = xb[m * XS_STR + col];
            const float xz = xb[m * XS_STR + H_DIM + col];
            const float xn = xb[m * XS_STR + 2 * H_DIM + col];

            const float r  = 1.0f / (1.0f + expf(-(xr + ar[v] + br)));
            const float z  = 1.0f / (1.0f + expf(-(xz + az[v] + bz)));
            const float nn = tanhf(xn + r * (an[v] + bnn));
            const float hn = (1.0f - z) * nn + z * hreg[v];

            hreg[v] = hn;
            Hs[m * LDS_STR + col] = hn;
            __builtin_nontemporal_store(
                hn, &out_seq[((size_t)(b0 + m) * T_LEN + t) * H_DIM + col]);
        }

        __syncthreads();   // new h visible before next step's A-frag reads
    }
}

// ---------------------------------------------------------------------------
// Phase 3: out = sigmoid(seq[:, T-1, :] @ W_fc^T + b_fc)   (128 x 256)
// ---------------------------------------------------------------------------
__global__ __launch_bounds__(256)
void fc_kernel(const float* __restrict__ seq,   // (B, T, 256)
               const float* __restrict__ Wfc,   // (256, 256)
               const float* __restrict__ bfc,   // (256)
               float* __restrict__ out)         // (128, 256)
{
    const int tid  = threadIdx.x;
    const int wv   = tid >> 5;
    const int lane = tid & 31;
    const int lm   = lane & 15;
    const int lk2  = (lane >> 4) * 2;
    const int m0   = blockIdx.x * 16;
    const int n0   = (blockIdx.y * 8 + wv) * 16;

    const float* arow = seq + ((size_t)(m0 + lm) * T_LEN + (T_LEN - 1)) * H_DIM + lk2;
    const float* brow = Wfc + (size_t)(n0 + lm) * H_DIM + lk2;

    v8f acc = {};
#pragma unroll 8
    for (int k = 0; k < H_DIM; k += 4) {
        v2f a = *(const v2f*)(arow + k);
        v2f b = *(const v2f*)(brow + k);
        acc = wmma_f32(a, b, acc);
    }

    const float bn = bfc[n0 + lm];
    const int   mb = (lane >> 4) * 8;
#pragma unroll
    for (int v = 0; v < 8; ++v) {
        const float s = 1.0f / (1.0f + expf(-(acc[v] + bn)));
        out[(size_t)(m0 + mb + v) * H_DIM + n0 + lm] = s;
    }
}

// ---------------------------------------------------------------------------
// Inputs (setup_inputs order):
//  0: x (128,1024,256)  1: W_ih (2,768,256)  2: W_hh (2,768,256)
//  3: b_ih (2,768)      4: b_hh (2,768)      5: W_fc (256,256)   6: b_fc (256)
// ---------------------------------------------------------------------------
extern "C" void kernel_launch(void* const* d_in, const int* in_sizes, int n_in,
                              void* d_out, int out_size, void* d_ws, size_t ws_size,
                              hipStream_t stream) {
    const float* x   = (const float*)d_in[0];
    const float* Wih = (const float*)d_in[1];
    const float* Whh = (const float*)d_in[2];
    const float* bih = (const float*)d_in[3];
    const float* bhh = (const float*)d_in[4];
    const float* Wfc = (const float*)d_in[5];
    const float* bfc = (const float*)d_in[6];
    float* out = (float*)d_out;

    float* XG  = (float*)d_ws;                               // B*T*768 f32 (402.7 MB)
    float* SEQ = XG + (size_t)B_SZ * T_LEN * G3;             // B*T*256 f32 (134.2 MB)

    const dim3 gGemm(B_SZ * T_LEN / 16, G3 / (16 * 8));      // (8192, 6)
    const dim3 gScan(B_SZ / 16);                             // 8 independent WGs
    const dim3 gFc(B_SZ / 16, H_DIM / (16 * 8));             // (8, 2)

    // Layer 0
    xg_gemm_kernel<<<gGemm, 256, 0, stream>>>(x, Wih, bih, XG);
    gru_scan_kernel<<<gScan, 512, 0, stream>>>(XG, Whh, bhh, SEQ);
    // Layer 1 (XG reused)
    xg_gemm_kernel<<<gGemm, 256, 0, stream>>>(SEQ, Wih + (size_t)G3 * H_DIM,
                                              bih + G3, XG);
    gru_scan_kernel<<<gScan, 512, 0, stream>>>(XG, Whh + (size_t)G3 * H_DIM,
                                               bhh + G3, SEQ);
    // FC head
    fc_kernel<<<gFc, 256, 0, stream>>>(SEQ, Wfc, bfc, out);
}